// GCN_32444182954421
// MI455X (gfx1250) — compile-verified
//
#include <hip/hip_runtime.h>

typedef float v2f __attribute__((ext_vector_type(2)));
typedef float v8f __attribute__((ext_vector_type(8)));
typedef float f4v __attribute__((ext_vector_type(4)));
typedef int   i2v __attribute__((ext_vector_type(2)));

#define MAXG 128

// ---------------------------------------------------------------------------
// Kernel 0: zero the workspace accumulators (must be re-done every call since
// the harness replays the graph without re-poisoning).
// ---------------------------------------------------------------------------
__global__ void gcn_zero_ws(float* __restrict__ ws, long long nwords) {
    long long i = (long long)blockIdx.x * blockDim.x + threadIdx.x;
    if (i < nwords) ws[i] = 0.0f;
}

// ---------------------------------------------------------------------------
// Kernel 1: edge scatter. 2 edges per thread -> 16B polar load + 8B dst load.
// Non-temporal loads keep the 154MB edge stream from evicting the 2.4MB of
// node accumulators living in L2. 3 L2 atomics per edge.
// ---------------------------------------------------------------------------
__global__ void gcn_edge_scatter(const f4v* __restrict__ polar4,
                                 const i2v* __restrict__ dst2,
                                 float* __restrict__ segx,
                                 float* __restrict__ segy,
                                 unsigned* __restrict__ deg,
                                 long long E) {
    long long t    = (long long)blockIdx.x * blockDim.x + threadIdx.x;
    long long base = t * 2;
    if (base + 1 < E) {
        f4v p = __builtin_nontemporal_load(&polar4[t]);
        i2v d = __builtin_nontemporal_load(&dst2[t]);
        atomicAdd(&segx[d.x], p.x);
        atomicAdd(&segy[d.x], p.y);
        atomicAdd(&deg[d.x], 1u);
        atomicAdd(&segx[d.y], p.z);
        atomicAdd(&segy[d.y], p.w);
        atomicAdd(&deg[d.y], 1u);
    } else if (base < E) {  // odd-E tail
        const v2f* polar2 = (const v2f*)polar4;
        const int* dst    = (const int*)dst2;
        v2f p = __builtin_nontemporal_load(&polar2[base]);
        int d = dst[base];
        atomicAdd(&segx[d], p.x);
        atomicAdd(&segy[d], p.y);
        atomicAdd(&deg[d], 1u);
    }
}

// ---------------------------------------------------------------------------
// Kernel 2: per-node h = relu(seg/deg), reduced per-graph. LDS (ds_add_f32)
// pre-reduction; graph_ids are sorted so each block flushes only the ~2
// graphs it touched -> ~2k global atomics total instead of 600k.
// ---------------------------------------------------------------------------
__global__ void gcn_node_reduce(const float* __restrict__ segx,
                                const float* __restrict__ segy,
                                const unsigned* __restrict__ deg,
                                const int* __restrict__ gid,
                                float* __restrict__ gacc,  // [gx[G] | gy[G] | cnt[G]]
                                int N, int G) {
    __shared__ float s_gx[MAXG], s_gy[MAXG], s_cnt[MAXG];
    for (int g = threadIdx.x; g < G; g += blockDim.x) {
        s_gx[g] = 0.0f; s_gy[g] = 0.0f; s_cnt[g] = 0.0f;
    }
    __syncthreads();

    int n = blockIdx.x * blockDim.x + threadIdx.x;
    if (n < N) {
        unsigned dg = deg[n];
        float hx = 0.0f, hy = 0.0f;
        if (dg > 0u) {
            float inv = 1.0f / (float)dg;
            hx = fmaxf(segx[n] * inv, 0.0f);
            hy = fmaxf(segy[n] * inv, 0.0f);
        }
        int g = gid[n];
        atomicAdd(&s_gx[g], hx);
        atomicAdd(&s_gy[g], hy);
        atomicAdd(&s_cnt[g], 1.0f);
    }
    __syncthreads();

    for (int g = threadIdx.x; g < G; g += blockDim.x) {
        float c = s_cnt[g];
        if (c != 0.0f) {  // sorted graph_ids -> most slots are empty
            atomicAdd(&gacc[g],         s_gx[g]);
            atomicAdd(&gacc[G + g],     s_gy[g]);
            atomicAdd(&gacc[2 * G + g], c);
        }
    }
}

// ---------------------------------------------------------------------------
// Kernel 3: hg = gsum/max(cnt,1);  out = hg @ W + b  via V_WMMA_F32_16X16X4_F32.
// [64,2]@[2,10]: 4 tiles over M, K padded 2->4 (zeros), N padded 10->16.
// A 16x4 f32: lanes 0-15 hold {K0,K1} for M=lane; lanes 16-31 hold the K=2,3
// zero pad. B 4x16 mirrored (lanes 0-15 = W rows 0,1 at N=lane). One wave,
// EXEC all-ones as required.
// ---------------------------------------------------------------------------
__global__ void gcn_final_wmma(const float* __restrict__ gacc,
                               const float* __restrict__ W,   // [2,C]
                               const float* __restrict__ bias,// [C]
                               float* __restrict__ out,       // [G,C]
                               int G, int C) {
    int  lane = threadIdx.x;
    int  col  = lane & 15;
    bool lo   = lane < 16;

    v2f Bf;
    Bf.x = (lo && col < C) ? W[col]     : 0.0f;  // B[K=0][col]
    Bf.y = (lo && col < C) ? W[C + col] : 0.0f;  // B[K=1][col]
    float bb = (col < C) ? bias[col] : 0.0f;

    int ntiles = (G + 15) >> 4;
    for (int t = 0; t < ntiles; ++t) {
        int   row = t * 16 + lane;  // meaningful for lanes 0-15
        float a0 = 0.0f, a1 = 0.0f;
        if (lo && row < G) {
            float cnt = gacc[2 * G + row];
            float inv = 1.0f / fmaxf(cnt, 1.0f);
            a0 = gacc[row] * inv;       // hg[row][0]
            a1 = gacc[G + row] * inv;   // hg[row][1]
        }
        v2f Af; Af.x = a0; Af.y = a1;

        v8f Cm = {};
        Cm = __builtin_amdgcn_wmma_f32_16x16x4_f32(
            /*neg_a=*/false, Af, /*neg_b=*/false, Bf,
            /*c_mod=*/(short)0, Cm, /*reuse_a=*/false, /*reuse_b=*/false);

        if (col < C) {
            int rbase = t * 16 + (lo ? 0 : 8);  // C/D layout: VGPR r -> M=r (+8 upper half)
#pragma unroll
            for (int r = 0; r < 8; ++r) {
                int orow = rbase + r;
                if (orow < G) out[orow * C + col] = Cm[r] + bb;
            }
        }
    }
}

// ---------------------------------------------------------------------------
extern "C" void kernel_launch(void* const* d_in, const int* in_sizes, int n_in,
                              void* d_out, int out_size, void* d_ws, size_t ws_size,
                              hipStream_t stream) {
    const float* polar = (const float*)d_in[0];  // [E,2]
    const int*   dst   = (const int*)  d_in[1];  // [E]
    const int*   gid   = (const int*)  d_in[2];  // [N]
    const float* W     = (const float*)d_in[3];  // [2,C]
    const float* bias  = (const float*)d_in[4];  // [C]

    long long E = in_sizes[1];
    int       N = in_sizes[2];
    int       C = in_sizes[4];
    int       G = out_size / C;

    float*    segx = (float*)d_ws;
    float*    segy = segx + N;
    unsigned* deg  = (unsigned*)(segy + N);
    float*    gacc = (float*)(deg + N);

    // 0) zero accumulators
    long long zwords = 3LL * N + 3LL * G;
    gcn_zero_ws<<<(int)((zwords + 255) / 256), 256, 0, stream>>>((float*)d_ws, zwords);

    // 1) edge scatter (bandwidth-bound: ~154MB streamed)
    long long pairs = (E + 1) / 2;
    gcn_edge_scatter<<<(int)((pairs + 255) / 256), 256, 0, stream>>>(
        (const f4v*)polar, (const i2v*)dst, segx, segy, deg, E);

    // 2) relu(seg/deg) + per-graph LDS-reduced sums
    gcn_node_reduce<<<(N + 255) / 256, 256, 0, stream>>>(segx, segy, deg, gid, gacc, N, G);

    // 3) tiny GEMM on tensor path (single wave, 4x v_wmma_f32_16x16x4_f32)
    gcn_final_wmma<<<1, 32, 0, stream>>>(gacc, W, bias, (float*)d_out, G, C);
}